// GaussianKernel_7378753814837
// MI455X (gfx1250) — compile-verified
//
#include <hip/hip_runtime.h>
#include <hip/hip_bf16.h>

#define Nb 32
#define Rr 128
#define Ff 128

typedef float fv2 __attribute__((ext_vector_type(2)));
typedef float fv4 __attribute__((ext_vector_type(4)));
typedef float fv8 __attribute__((ext_vector_type(8)));

// One block per (batch n, 16-row i-tile). 256 threads = 8 waves (wave32).
// Wave w computes the 16x16 WMMA tile at columns [16w, 16w+16) using
// V_WMMA_F32_16X16X4_F32 (full fp32 matrix pipe), then the block does the
// row softmax out of LDS.
__global__ __launch_bounds__(256)
void gaussian_softmax_wmma(const float* __restrict__ x1,
                           const float* __restrict__ x2,
                           const float* __restrict__ sigma_p,
                           const float* __restrict__ mean_p,
                           float* __restrict__ out) {
    __shared__ float sc_lds[Rr];       // ||x2[j,:] + mean||^2 for all 128 j
    __shared__ float sa_lds[16];       // ||x1[i,:]||^2 for this tile's rows
    __shared__ float tile[16 * Rr];    // G strip: 16 rows x 128 cols

    const int tid   = threadIdx.x;
    const int n     = blockIdx.y;
    const int ibase = blockIdx.x * 16;

    const float mean  = mean_p[0];
    const float sigma = sigma_p[0];

    // ---- Phase 1: row norms (cooperative) ----
    if (tid < Rr) {
        const float* p = x2 + (size_t)(n * Rr + tid) * Ff;
        float s = 0.f;
#pragma unroll 4
        for (int f = 0; f < Ff; f += 4) {
            fv4 v = *(const fv4*)(p + f);
            float a0 = v.x + mean, a1 = v.y + mean;
            float a2 = v.z + mean, a3 = v.w + mean;
            s += a0 * a0 + a1 * a1 + a2 * a2 + a3 * a3;
        }
        sc_lds[tid] = s;
    } else if (tid < Rr + 16) {
        const int ii = tid - Rr;
        const float* p = x1 + (size_t)(n * Rr + ibase + ii) * Ff;
        float s = 0.f;
#pragma unroll 4
        for (int f = 0; f < Ff; f += 4) {
            fv4 v = *(const fv4*)(p + f);
            s += v.x * v.x + v.y * v.y + v.z * v.z + v.w * v.w;
        }
        sa_lds[ii] = s;
    }
    __syncthreads();

    // ---- Phase 2: fp32 WMMA GEMM, G = X1_tile * (X2 + mean)^T ----
    const int lane  = tid & 31;
    const int w     = tid >> 5;     // wave id -> j-tile
    const int lo    = lane & 15;
    const int hi    = lane >> 4;
    const int jbase = w * 16;

    // A (16x4 f32): lane holds row M=lo, K-pair {2*hi, 2*hi+1}
    // B (4x16 f32): lane holds col N=lo, K-pair {2*hi, 2*hi+1}; B[k,j]=C[j,k]
    const float* aptr = x1 + (size_t)(n * Rr + ibase + lo) * Ff + 2 * hi;
    const float* bptr = x2 + (size_t)(n * Rr + jbase + lo) * Ff + 2 * hi;

    fv8 acc = {};
#pragma unroll 4
    for (int k = 0; k < Ff; k += 4) {
        fv2 a = *(const fv2*)(aptr + k);
        fv2 c = *(const fv2*)(bptr + k);
        fv2 b;
        b.x = c.x + mean;
        b.y = c.y + mean;
        acc = __builtin_amdgcn_wmma_f32_16x16x4_f32(
            /*neg_a=*/false, a, /*neg_b=*/false, b,
            /*c_mod=*/(short)0, acc, /*reuse_a=*/false, /*reuse_b=*/false);
    }

    // ---- Phase 3: D-matrix layout -> LDS strip ----
    // lane l, VGPR r : M = r + 8*(l>=16), N = l%16
#pragma unroll
    for (int r = 0; r < 8; ++r) {
        tile[(r + 8 * hi) * Rr + jbase + lo] = acc[r];
    }
    __syncthreads();

    // ---- Phase 4: exp kernel value + row softmax ----
    // 16 threads per row; rows 2w and 2w+1 live in wave w's two 16-lane halves
    const int row = (w << 1) + hi;
    const int c0  = lo;
    const float inv2s2 = 1.0f / (2.0f * sigma * sigma);
    const float sa = sa_lds[row];

    float kv[8];
    float mx = -3.402823466e38f;
#pragma unroll
    for (int q = 0; q < 8; ++q) {
        const int j = c0 + (q << 4);
        const float g  = tile[row * Rr + j];
        const float d2 = sa + sc_lds[j] - 2.0f * g;
        const float v  = __expf(-d2 * inv2s2);
        kv[q] = v;
        mx = fmaxf(mx, v);
    }
    // max over the 16 lanes of this row (xor masks < 16 stay in the half-wave)
#pragma unroll
    for (int off = 1; off < 16; off <<= 1)
        mx = fmaxf(mx, __shfl_xor(mx, off, 32));

    float ev[8];
    float sum = 0.f;
#pragma unroll
    for (int q = 0; q < 8; ++q) {
        ev[q] = __expf(kv[q] - mx);
        sum += ev[q];
    }
#pragma unroll
    for (int off = 1; off < 16; off <<= 1)
        sum += __shfl_xor(sum, off, 32);

    const float rs = 1.0f / sum;
    const size_t orow = ((size_t)n * Rr + (ibase + row)) * Rr;
#pragma unroll
    for (int q = 0; q < 8; ++q) {
        out[orow + c0 + (q << 4)] = ev[q] * rs;
    }
}

extern "C" void kernel_launch(void* const* d_in, const int* in_sizes, int n_in,
                              void* d_out, int out_size, void* d_ws, size_t ws_size,
                              hipStream_t stream) {
    (void)in_sizes; (void)n_in; (void)out_size; (void)d_ws; (void)ws_size;
    const float* x1    = (const float*)d_in[0];
    const float* x2    = (const float*)d_in[1];
    const float* sigma = (const float*)d_in[2];
    const float* mean  = (const float*)d_in[3];
    float* out = (float*)d_out;

    dim3 grid(Rr / 16, Nb, 1);   // 8 i-tiles x 32 batches
    dim3 block(256, 1, 1);       // 8 waves
    gaussian_softmax_wmma<<<grid, block, 0, stream>>>(x1, x2, sigma, mean, out);
}